// GPLayer_26027501814505
// MI455X (gfx1250) — compile-verified
//
#include <hip/hip_runtime.h>
#include <stdint.h>

#define DFEAT 64
#define EPC 32              // edges per chunk == wave32 lanes
#define WPB 8               // waves per block (256 threads)
#define BLOCK 256
#define GRID_SPMM 1024

// ---------------------------------------------------------------------------
// Zero the output (harness poisons d_out; we accumulate with atomics).
// ---------------------------------------------------------------------------
__global__ __launch_bounds__(BLOCK) void zero_f4(float4* __restrict__ o, int n4) {
  int i = blockIdx.x * BLOCK + threadIdx.x;
  if (i < n4) o[i] = float4{0.f, 0.f, 0.f, 0.f};
}

// ---------------------------------------------------------------------------
// COO SpMM: out[row] += val * features[col]
//  - one wave per 32-edge chunk (grid-strided)
//  - edge triples staged into LDS via CDNA5 async global->LDS loads,
//    double-buffered per wave, synchronized with s_wait_asynccnt (ASYNCcnt)
//  - readlane broadcasts (row, col, val) -> SGPRs so feature loads and
//    atomics use the scalar-base (saddr) form
//  - 32 lanes x float2 covers one 64-float feature row (2x128B transactions)
//  - scatter via non-returning global_atomic_add_f32 scope:SCOPE_DEV
//    (RMW resolves in L2 atomic units; out fits in the 192MB L2)
//  - atomic asm carries NO memory clobber so the unrolled loop can keep
//    several feature-row gathers in flight past outstanding atomics
// ---------------------------------------------------------------------------
__global__ __launch_bounds__(BLOCK) void spmm_coo_atomic(
    const float* __restrict__ feat,
    const int*   __restrict__ rows,
    const int*   __restrict__ cols,
    const float* __restrict__ vals,
    float*       __restrict__ out,
    int nnz)
{
  __shared__ int   s_rows[WPB][2][EPC];
  __shared__ int   s_cols[WPB][2][EPC];
  __shared__ float s_vals[WPB][2][EPC];

  const int lane = threadIdx.x & 31;
  const int wv   = threadIdx.x >> 5;
  const int wid  = blockIdx.x * WPB + wv;
  const int nw   = gridDim.x * WPB;
  const int nchunks = (nnz + EPC - 1) / EPC;

  const uint64_t rows64 = (uint64_t)(uintptr_t)rows;
  const uint64_t cols64 = (uint64_t)(uintptr_t)cols;
  const uint64_t vals64 = (uint64_t)(uintptr_t)vals;
  const uint64_t out64  = (uint64_t)(uintptr_t)out;

  // Stage chunk c's 32 edge triples into per-wave LDS buffer b (async).
  auto issue = [&](int c, int b) {
    int e = c * EPC + lane;
    const int emax = nnz - 1;
    if (e > emax) e = emax;                       // clamp tail (garbage lanes never consumed)
    const uint32_t voff = (uint32_t)e << 2;       // byte offset into int/float arrays
    // generic->LDS address: low 32 bits of the flat pointer are the LDS offset
    const uint32_t l_r = (uint32_t)(uintptr_t)&s_rows[wv][b][lane];
    const uint32_t l_c = (uint32_t)(uintptr_t)&s_cols[wv][b][lane];
    const uint32_t l_v = (uint32_t)(uintptr_t)&s_vals[wv][b][lane];
    asm volatile("global_load_async_to_lds_b32 %0, %1, %2"
                 :: "v"(l_r), "v"(voff), "s"(rows64) : "memory");
    asm volatile("global_load_async_to_lds_b32 %0, %1, %2"
                 :: "v"(l_c), "v"(voff), "s"(cols64) : "memory");
    asm volatile("global_load_async_to_lds_b32 %0, %1, %2"
                 :: "v"(l_v), "v"(voff), "s"(vals64) : "memory");
  };

  // Process one edge (row/col/val broadcast from lane j of this chunk).
  auto edge = [&](int r, int k, float v, int j) {
    const int   rj = __builtin_amdgcn_readlane(r, j);
    const int   kj = __builtin_amdgcn_readlane(k, j);
    const float vj = __uint_as_float(
        (unsigned)__builtin_amdgcn_readlane((int)__float_as_uint(v), j));

    // Gather: 32 lanes x float2 = one 64-float feature row (saddr-form b64 load).
    const float2 f = *((const float2*)(feat + (size_t)kj * DFEAT) + lane);
    const float mx = f.x * vj;
    const float my = f.y * vj;

    // Scatter-add: two non-returning f32 atomics per lane, scalar base + voffset.
    // volatile keeps them, no memory clobber so independent loads can overlap.
    const uint32_t ooff = (((uint32_t)rj * DFEAT) + (uint32_t)(lane << 1)) << 2;
    asm volatile("global_atomic_add_f32 %0, %1, %2 scope:SCOPE_DEV"
                 :: "v"(ooff), "v"(mx), "s"(out64));
    asm volatile("global_atomic_add_f32 %0, %1, %2 offset:4 scope:SCOPE_DEV"
                 :: "v"(ooff), "v"(my), "s"(out64));
  };

  int c = wid;
  if (c >= nchunks) return;

  issue(c, 0);                                    // prologue: fill buffer 0
  int buf = 0;

  for (; c < nchunks; c += nw, buf ^= 1) {
    const int  cn  = c + nw;
    const bool pre = (cn < nchunks);
    if (pre) issue(cn, buf ^ 1);                  // prefetch next chunk into other buffer

    // Async loads complete in order: with a prefetch in flight, the 3 newest
    // outstanding ops belong to chunk cn -> wait until only those remain.
    if (pre) asm volatile("s_wait_asynccnt 0x3" ::: "memory");
    else     asm volatile("s_wait_asynccnt 0x0" ::: "memory");

    const int   r = s_rows[wv][buf][lane];
    const int   k = s_cols[wv][buf][lane];
    const float v = s_vals[wv][buf][lane];

    int cnt = nnz - c * EPC;
    if (cnt > EPC) cnt = EPC;

    if (cnt == EPC) {
      // Common path: full chunk, unrolled so multiple row gathers stay in flight.
      #pragma unroll 8
      for (int j = 0; j < EPC; ++j) edge(r, k, v, j);
    } else {
      for (int j = 0; j < cnt; ++j) edge(r, k, v, j);
    }
  }
}

// ---------------------------------------------------------------------------
// Harness entry
// ---------------------------------------------------------------------------
extern "C" void kernel_launch(void* const* d_in, const int* in_sizes, int n_in,
                              void* d_out, int out_size, void* d_ws, size_t ws_size,
                              hipStream_t stream) {
  const float* feat = (const float*)d_in[0];   // [N, 64] f32
  const int*   rows = (const int*)d_in[1];     // [nnz] i32
  const int*   cols = (const int*)d_in[2];     // [nnz] i32
  const float* vals = (const float*)d_in[3];   // [nnz] f32
  float* out = (float*)d_out;                  // [N, 64] f32
  (void)n_in; (void)d_ws; (void)ws_size;

  const int nnz = in_sizes[1];

  // 1) zero the accumulation target (b128 stores)
  const int n4 = out_size / 4;
  zero_f4<<<(n4 + BLOCK - 1) / BLOCK, BLOCK, 0, stream>>>((float4*)d_out, n4);

  // 2) scatter-gather SpMM
  const int nchunks = (nnz + EPC - 1) / EPC;
  int grid = GRID_SPMM;
  const int max_grid = (nchunks + WPB - 1) / WPB;
  if (grid > max_grid) grid = max_grid;
  if (grid < 1) grid = 1;
  spmm_coo_atomic<<<grid, BLOCK, 0, stream>>>(feat, rows, cols, vals, out, nnz);
}